// SpikingInsularCortex_21294447854015
// MI455X (gfx1250) — compile-verified
//
#include <hip/hip_runtime.h>
#include <hip/hip_bf16.h>
#include <math.h>

// ---------------------------------------------------------------------------
// Sizes (match reference)
// ---------------------------------------------------------------------------
#define N_NEURONS 1048576
#define T_STEPS   15
#define NEU_PER_THREAD 4
#define BLOCK     256
#define NEU_PER_BLOCK (BLOCK * NEU_PER_THREAD)        // 1024 neurons / block
#define BLOCKS_PER_CH (N_NEURONS / NEU_PER_BLOCK)     // 1024 blocks / channel

typedef float v2f __attribute__((ext_vector_type(2)));
typedef float v8f __attribute__((ext_vector_type(8)));

// ---------------------------------------------------------------------------
// Exact-precision wave32 reduction via V_WMMA_F32_16X16X4_F32.
// A (16x4 f32): lane L<16 holds {A[L][0],A[L][1]}, lane L>=16 holds
// {A[L-16][2],A[L-16][3]}. With a = {p, 0}, the 32 partials occupy K=0 and
// K=2; B = all-ones 4x16 makes D[m][n] = rowsum_m (every column identical).
// D layout: VGPR r, lanes 0-15 -> M=r; lanes 16-31 -> M=8+r. So summing a
// lane's 8 D regs gives rows 0..7 (lanes<16) or rows 8..15 (lanes>=16).
// ---------------------------------------------------------------------------
__device__ __forceinline__ float wave_reduce_wmma(float p) {
    v2f a; a.x = p;    a.y = 0.0f;
    v2f b; b.x = 1.0f; b.y = 1.0f;
    v8f c = {};
    c = __builtin_amdgcn_wmma_f32_16x16x4_f32(
            /*neg_a=*/false, a, /*neg_b=*/false, b,
            /*c_mod=*/(short)0, c, /*reuse_a=*/false, /*reuse_b=*/false);
    float h = c[0] + c[1] + c[2] + c[3] + c[4] + c[5] + c[6] + c[7];
    return __shfl(h, 0, 32) + __shfl(h, 16, 32);
}

// One Izhikevich Euler step (matches reference ordering exactly).
__device__ __forceinline__ void izstep(float& v, float& u, float& r, float Iin) {
    float vn = v + (0.04f * v * v + 5.0f * v + 140.0f - u + Iin);
    float un = u + 0.02f * (0.2f * v - u);
    float spk = (vn >= 30.0f) ? 1.0f : 0.0f;
    v = (spk > 0.0f) ? -65.0f : vn;
    u = un + spk * 8.0f;
    r = 0.9f * r + spk;
}

// ---------------------------------------------------------------------------
// Main streaming kernel: grid = (1024, 3), block = 256, 4 neurons/thread.
// Reads ~239 MB once; memory-bound; all 15 noise loads issued up front for
// maximum memory-level parallelism, recurrence runs from registers.
// ---------------------------------------------------------------------------
__global__ void __launch_bounds__(BLOCK)
izhikevich_scan_kernel(const float* __restrict__ energy_p,
                       const float* __restrict__ stress_p,
                       const float* __restrict__ fatigue_p,
                       const float* __restrict__ v0,
                       const float* __restrict__ u0,
                       const float* __restrict__ rate0,
                       const float* __restrict__ itonic,
                       const float* __restrict__ noise,
                       float* __restrict__ partials) {
    const int c    = blockIdx.y;
    const int base = blockIdx.x * NEU_PER_BLOCK + threadIdx.x * NEU_PER_THREAD;
    const size_t g = (size_t)c * N_NEURONS + base;

    // Per-channel drive current I = |eps[c]| * GAIN (scalar broadcast loads).
    const float e  = *energy_p;
    const float st = *stress_p;
    const float fa = *fatigue_p;
    float eps;
    if (c == 0)      eps = fmaxf(0.0f, 1.0f - e * 0.01f) - 0.25f;
    else if (c == 1) eps = fa - 0.20f;
    else             eps = st - 0.10f;
    const float I = fabsf(eps) * 15.0f;

    float4 v  = *(const float4*)(v0     + g);
    float4 u  = *(const float4*)(u0     + g);
    float4 r  = *(const float4*)(rate0  + g);
    float4 it = *(const float4*)(itonic + g);

    // Pre-issue all T noise loads (b128 each) — MLP to hide HBM latency.
    float4 nz[T_STEPS];
#pragma unroll
    for (int t = 0; t < T_STEPS; ++t)
        nz[t] = *(const float4*)(noise + (size_t)t * (3u * N_NEURONS) + g);

    const float bias = I;  // + per-neuron i_tonic below
#pragma unroll
    for (int t = 0; t < T_STEPS; ++t) {
        izstep(v.x, u.x, r.x, bias + nz[t].x + it.x);
        izstep(v.y, u.y, r.y, bias + nz[t].y + it.y);
        izstep(v.z, u.z, r.z, bias + nz[t].z + it.z);
        izstep(v.w, u.w, r.w, bias + nz[t].w + it.w);
    }

    // Thread partial -> WMMA wave reduction -> LDS -> one value per block.
    float p = r.x + r.y + r.z + r.w;
    float wsum = wave_reduce_wmma(p);   // EXEC all ones here (no divergence)

    __shared__ float lds[BLOCK / 32];
    const int wave = threadIdx.x >> 5;
    if ((threadIdx.x & 31) == 0) lds[wave] = wsum;
    __syncthreads();
    if (threadIdx.x == 0) {
        float bsum = 0.0f;
#pragma unroll
        for (int w = 0; w < BLOCK / 32; ++w) bsum += lds[w];
        partials[c * BLOCKS_PER_CH + blockIdx.x] = bsum;
    }
}

// ---------------------------------------------------------------------------
// Deterministic fixed-order final reduction + scalar affect math (1 block).
// ---------------------------------------------------------------------------
__global__ void __launch_bounds__(256)
finalize_kernel(const float* __restrict__ partials,
                const float* __restrict__ energy_p,
                const float* __restrict__ stress_p,
                const float* __restrict__ fatigue_p,
                const float* __restrict__ reward_p,
                const int*   __restrict__ threat_p,
                float* __restrict__ out) {
    __shared__ float red[256];
    float sums[3];

    for (int c = 0; c < 3; ++c) {
        float s = 0.0f;
        for (int i = threadIdx.x; i < BLOCKS_PER_CH; i += 256)
            s += partials[c * BLOCKS_PER_CH + i];
        red[threadIdx.x] = s;
        __syncthreads();
        for (int off = 128; off > 0; off >>= 1) {
            if ((int)threadIdx.x < off) red[threadIdx.x] += red[threadIdx.x + off];
            __syncthreads();
        }
        sums[c] = red[0];
        __syncthreads();
    }

    if (threadIdx.x == 0) {
        const float invN    = 1.0f / (float)N_NEURONS;
        const float e       = *energy_p;
        const float stress  = *stress_p;
        const float fatigue = *fatigue_p;
        const float reward  = *reward_p;
        const int   threat_acute = *threat_p;

        const float setp[3]  = {0.25f, 0.20f, 0.10f};
        const float pi_[3]   = {2.0f, 1.5f, 2.5f};
        const float vsign[3] = {1.0f, 0.6f, 1.0f};

        float actual[3];
        actual[0] = fmaxf(0.0f, 1.0f - e * 0.01f);
        actual[1] = fatigue;
        actual[2] = stress;

        float eps[3];
        float raw_valence = 0.0f, raw_arousal = 0.0f;
        for (int c = 0; c < 3; ++c) {
            eps[c] = actual[c] - setp[c];
            raw_valence -= vsign[c] * pi_[c] * eps[c];
            raw_arousal += pi_[c] * eps[c] * eps[c];
        }
        raw_valence += reward * 1.5f;

        float valence = fminf(1.0f, fmaxf(-1.0f, 0.15f * raw_valence));
        float arousal = fminf(1.0f, 0.1f * raw_arousal);

        float hr_target = 2.0f + 4.0f * arousal + 1.0f * stress;
        bool threat = (threat_acute != 0) && (stress > 0.5f);
        if (threat) hr_target = fmaxf(1.0f, hr_target * 0.5f);
        if ((fatigue > 0.3f) && (stress > 0.3f))
            hr_target = fminf(8.0f, hr_target * 1.3f);
        float heart_rate = fminf(8.0f, fmaxf(0.5f, 0.9f * 2.0f + 0.1f * hr_target));
        float heart_phase = heart_rate * 0.05f * 2.0f * 3.14159265358979323846f;
        float heartbeat = (sinf(heart_phase) > 0.9f) ? 1.0f : 0.0f;

        out[0] = sums[0] * invN;
        out[1] = sums[1] * invN;
        out[2] = sums[2] * invN;
        out[3] = valence;
        out[4] = arousal;
        out[5] = heart_rate;
        out[6] = heartbeat;
    }
}

// ---------------------------------------------------------------------------
// Launch
// ---------------------------------------------------------------------------
extern "C" void kernel_launch(void* const* d_in, const int* in_sizes, int n_in,
                              void* d_out, int out_size, void* d_ws, size_t ws_size,
                              hipStream_t stream) {
    // setup_inputs() order:
    // 0 energy, 1 stress, 2 fatigue, 3 reward, 4 threat_acute,
    // 5 v0, 6 u0, 7 rate0, 8 i_tonic, 9 noise
    const float* energy  = (const float*)d_in[0];
    const float* stress  = (const float*)d_in[1];
    const float* fatigue = (const float*)d_in[2];
    const float* reward  = (const float*)d_in[3];
    const int*   threat  = (const int*)  d_in[4];
    const float* v0      = (const float*)d_in[5];
    const float* u0      = (const float*)d_in[6];
    const float* rate0   = (const float*)d_in[7];
    const float* itonic  = (const float*)d_in[8];
    const float* noise   = (const float*)d_in[9];
    float* out = (float*)d_out;

    float* partials = (float*)d_ws;   // 3 * 1024 floats = 12 KB

    dim3 grid(BLOCKS_PER_CH, 3, 1);
    dim3 block(BLOCK, 1, 1);
    izhikevich_scan_kernel<<<grid, block, 0, stream>>>(
        energy, stress, fatigue, v0, u0, rate0, itonic, noise, partials);

    finalize_kernel<<<dim3(1), dim3(256), 0, stream>>>(
        partials, energy, stress, fatigue, reward, threat, out);
}